// DITRInjector_73400991088931
// MI455X (gfx1250) — compile-verified
//
#include <hip/hip_runtime.h>
#include <hip/hip_bf16.h>
#include <math.h>

// Problem constants from the reference (fixed shapes).
#define NB 2
#define NV 6
#define NDIM 384
#define NPH 37
#define NPW 37
#define IMGW 518.0f
#define IMGH 518.0f
#define CH_STRIDE (NPH * NPW)            // 1369
#define VIEW_STRIDE (NDIM * CH_STRIDE)   // 525696
#define TBL_FLOATS (NB * 8 * 16)         // 256 floats = 1 KB

typedef __attribute__((ext_vector_type(2))) float v2f;
typedef __attribute__((ext_vector_type(8))) float v8f;

// ---------------------------------------------------------------------------
// Kernel 1: build coefficient table  tbl[b][vslot(8)][q(4)][k(4)]
//   q=0..2 : rows of P = K @ E[:3,:]   (3x4)
//   q=3    : depth row  E[2,:]
//   vslot >= NV : zero-filled (depth row 0 => depth<=0.1 => never valid)
// ---------------------------------------------------------------------------
__global__ void build_table_kernel(const float* __restrict__ intr,
                                   const float* __restrict__ extr,
                                   float* __restrict__ tbl) {
    int t = threadIdx.x;
    if (t >= NB * 8) return;
    int b = t >> 3, vs = t & 7;
    float row[16];
    if (vs < NV) {
        const float* K = intr + (b * NV + vs) * 9;
        const float* E = extr + (b * NV + vs) * 16;
        #pragma unroll
        for (int i = 0; i < 3; ++i)
            #pragma unroll
            for (int k = 0; k < 4; ++k) {
                float s = 0.0f;
                #pragma unroll
                for (int j = 0; j < 3; ++j) s += K[i * 3 + j] * E[j * 4 + k];
                row[i * 4 + k] = s;
            }
        #pragma unroll
        for (int k = 0; k < 4; ++k) row[12 + k] = E[8 + k];  // E[2][k]
    } else {
        #pragma unroll
        for (int k = 0; k < 16; ++k) row[k] = 0.0f;
    }
    #pragma unroll
    for (int k = 0; k < 16; ++k) tbl[t * 16 + k] = row[k];
}

// ---------------------------------------------------------------------------
// Projection of 16 points x 16 coefficient rows via V_WMMA_F32_16X16X4_F32.
//   A (16x4): M = 4*viewInGroup + q; K = homo component index.
//     lanes 0-15 : M=lane,    VGPR0=K0, VGPR1=K1
//     lanes16-31 : M=lane-16, VGPR0=K2, VGPR1=K3
//   B (4x16): N = point index.
//     VGPR0: lanes0-15 K=0 (x), lanes16-31 K=2 (z)
//     VGPR1: lanes0-15 K=1 (y), lanes16-31 K=3 (1.0)
//   D (16x16): VGPR r: lanes0-15 -> (M=r,   N=lane)
//                      lanes16-31-> (M=r+8, N=lane-16)
// ---------------------------------------------------------------------------
__device__ __forceinline__ v8f compute_D(const float* __restrict__ tbl,
                                         int bb, int g, int lane, v2f Bm,
                                         float x, float y, float z) {
    int m  = lane & 15;
    int hi = lane >> 4;
#if defined(__HIP_DEVICE_COMPILE__) && __has_builtin(__builtin_amdgcn_wmma_f32_16x16x4_f32)
    int q = m & 3;
    int vloc = m >> 2;
    int base = (((bb * 8) + (g * 4 + vloc)) * 4 + q) * 4 + (hi ? 2 : 0);
    v2f A;
    A.x = tbl[base + 0];
    A.y = tbl[base + 1];
    v8f c = {0.f, 0.f, 0.f, 0.f, 0.f, 0.f, 0.f, 0.f};
    return __builtin_amdgcn_wmma_f32_16x16x4_f32(false, A, false, Bm,
                                                 (short)0, c, false, false);
#else
    // Arithmetically identical VALU fallback (same D layout).
    v8f d;
    #pragma unroll
    for (int r = 0; r < 8; ++r) {
        int M = (hi ? 8 : 0) + r;
        int vloc = M >> 2, q = M & 3;
        int base = (((bb * 8) + (g * 4 + vloc)) * 4 + q) * 4;
        d[r] = tbl[base + 0] * x + tbl[base + 1] * y + tbl[base + 2] * z + tbl[base + 3];
    }
    return d;
#endif
}

__device__ __forceinline__ float fast_rcp(float x) {
#if defined(__HIP_DEVICE_COMPILE__)
    return __builtin_amdgcn_rcpf(x);   // v_rcp_f32, 1 ulp — plenty for binning
#else
    return 1.0f / x;
#endif
}

// Per-lane: evaluate the (up to) 2 views this lane holds; return packed best
// code (v<<16 | vp<<8 | up) or -1. Higher v wins via integer max.
__device__ __forceinline__ int best_code(v8f d, int g, int hi) {
    int best = -1;
    #pragma unroll
    for (int j = 0; j < 2; ++j) {
        int v = g * 4 + hi * 2 + j;
        if (v >= NV) continue;
        float px = d[j * 4 + 0], py = d[j * 4 + 1];
        float pz = d[j * 4 + 2], dep = d[j * 4 + 3];
        float rz = fast_rcp(pz);       // pz==0 -> Inf/NaN -> comparisons false
        float u  = px * rz;
        float vc = py * rz;
        bool valid = (dep > 0.1f) && (u >= 0.0f) && (u < IMGW) &&
                     (vc >= 0.0f) && (vc < IMGH);
        if (valid) {
            int up = (int)(u * (1.0f / 14.0f));   // u>=0: trunc == floor
            int vp = (int)(vc * (1.0f / 14.0f));
            up = up > NPW - 1 ? NPW - 1 : up;
            vp = vp > NPH - 1 ? NPH - 1 : vp;
            int code = (v << 16) | (vp << 8) | up;
            best = best > code ? best : code;
        }
    }
    return best;
}

// ---------------------------------------------------------------------------
// Kernel 2: wave32 handles 16 points; 4 WMMAs (2 batches x 2 view-groups);
// coefficient table staged to LDS via async global-to-LDS load; writes per-point
// gather base offset (or -1) to codes[].
// ---------------------------------------------------------------------------
__global__ __launch_bounds__(256) void proj_kernel(
        const float* __restrict__ points, const int* __restrict__ batch_idx,
        const float* __restrict__ tbl, int* __restrict__ codes, int N) {
    __shared__ float s_tbl[TBL_FLOATS];     // 256 floats: one per thread

    // Stage table into LDS with GLOBAL_LOAD_ASYNC_TO_LDS_B32 (ASYNCcnt path).
    // Flat address of an LDS object: low 32 bits == LDS byte offset.
#if defined(__HIP_DEVICE_COMPILE__)
    {
        unsigned ldsoff = (unsigned)(uintptr_t)&s_tbl[threadIdx.x];
        unsigned long long ga = (unsigned long long)(uintptr_t)(tbl + threadIdx.x);
        asm volatile("global_load_async_to_lds_b32 %0, %1, off"
                     :: "v"(ldsoff), "v"(ga) : "memory");
        asm volatile("s_wait_asynccnt 0" ::: "memory");
    }
#else
    s_tbl[threadIdx.x] = tbl[threadIdx.x];
#endif
    __syncthreads();                        // before any wave can early-exit

    int wave = blockIdx.x * (blockDim.x >> 5) + (threadIdx.x >> 5);
    int n0 = wave * 16;
    if (n0 >= N) return;                    // wave-uniform: EXEC stays full

    int lane = threadIdx.x & 31;
    int p = n0 + (lane & 15);
    int pc = p < N ? p : N - 1;             // clamp (keeps EXEC all-ones)
    float x = points[pc * 3 + 0];
    float y = points[pc * 3 + 1];
    float z = points[pc * 3 + 2];

    v2f Bm;
    Bm.x = (lane < 16) ? x : z;
    Bm.y = (lane < 16) ? y : 1.0f;

    v8f D00 = compute_D(s_tbl, 0, 0, lane, Bm, x, y, z);
    v8f D01 = compute_D(s_tbl, 0, 1, lane, Bm, x, y, z);
    v8f D10 = compute_D(s_tbl, 1, 0, lane, Bm, x, y, z);
    v8f D11 = compute_D(s_tbl, 1, 1, lane, Bm, x, y, z);

    int hi = lane >> 4;
    int c0 = max(best_code(D00, 0, hi), best_code(D01, 1, hi));
    int c1 = max(best_code(D10, 0, hi), best_code(D11, 1, hi));
    int bi = batch_idx[pc];
    int code = (bi == 0) ? c0 : c1;
    code = max(code, __shfl_xor(code, 16, 32)); // merge both half-lanes

    if (lane < 16 && p < N) {
        int outv = -1;
        if (code >= 0) {
            int v  = code >> 16;
            int vp = (code >> 8) & 0xFF;
            int up = code & 0xFF;
            outv = (bi * NV + v) * VIEW_STRIDE + vp * NPW + up;
        }
        codes[p] = outv;
    }
}

// ---------------------------------------------------------------------------
// Kernel 3: wave per point; lane owns 4 consecutive channels per group so
// stores are b128 (the 384 MB store stream is the HBM bottleneck); strided
// feature reads stay b32 and hit L2 (dino tensor is 25 MB << 192 MB L2).
// ---------------------------------------------------------------------------
__global__ __launch_bounds__(256) void gather_kernel(
        const float* __restrict__ dino, const int* __restrict__ codes,
        float* __restrict__ out, int N) {
    int n = blockIdx.x * (blockDim.x >> 5) + (threadIdx.x >> 5);
    if (n >= N) return;
    int lane = threadIdx.x & 31;
    int code = codes[n];
    float* o = out + (size_t)n * NDIM;
    if (code >= 0) {
        const float* src = dino + code;
        #pragma unroll
        for (int i = 0; i < NDIM / 128; ++i) {       // 3 groups of 128 ch
            int c0 = i * 128 + lane * 4;
            float4 val;
            val.x = src[(size_t)(c0 + 0) * CH_STRIDE];
            val.y = src[(size_t)(c0 + 1) * CH_STRIDE];
            val.z = src[(size_t)(c0 + 2) * CH_STRIDE];
            val.w = src[(size_t)(c0 + 3) * CH_STRIDE];
            *(float4*)(o + c0) = val;                // global_store_b128
        }
    } else {
        float4 zero = make_float4(0.f, 0.f, 0.f, 0.f);
        #pragma unroll
        for (int i = 0; i < NDIM / 128; ++i)
            *(float4*)(o + i * 128 + lane * 4) = zero;
    }
}

// ---------------------------------------------------------------------------
extern "C" void kernel_launch(void* const* d_in, const int* in_sizes, int n_in,
                              void* d_out, int out_size, void* d_ws, size_t ws_size,
                              hipStream_t stream) {
    const float* points = (const float*)d_in[0];
    // d_in[1] = imgs (unused by the reference output)
    const float* intr = (const float*)d_in[2];
    const float* extr = (const float*)d_in[3];
    const float* dino = (const float*)d_in[4];
    const int*   bidx = (const int*)d_in[5];
    int N = in_sizes[0] / 3;

    float* tbl  = (float*)d_ws;                     // 256 floats
    int*  codes = (int*)((char*)d_ws + 4096);       // N ints
    float* out  = (float*)d_out;

    build_table_kernel<<<1, 64, 0, stream>>>(intr, extr, tbl);

    int waves2  = (N + 15) / 16;
    int blocks2 = (waves2 + 7) / 8;
    proj_kernel<<<blocks2, 256, 0, stream>>>(points, bidx, tbl, codes, N);

    int blocks3 = (N + 7) / 8;
    gather_kernel<<<blocks3, 256, 0, stream>>>(dino, codes, out, N);
}